// CrossEntropyAndUnlikelihoodLoss_28054726377866
// MI455X (gfx1250) — compile-verified
//
#include <hip/hip_runtime.h>
#include <stdint.h>

// CrossEntropy + Unlikelihood loss, MI455X (gfx1250).
// HBM-bound streaming reduction: 512 MiB single pass, ~23 us roofline at 23.3 TB/s.
// Uses CDNA5 async global->LDS copies (ASYNCcnt) to double-buffer row fetches
// behind the barrier-heavy per-row reductions. No matmul in this op -> no WMMA.

#define IGNORE_INDEX (-100)
#define LOSS_C       1024          // classes (reference: C = 1024 = 4 floats * 256 threads)
#define LOSS_THREADS 256           // 8 waves (wave32)
#define MAX_BLOCKS   8192          // persistent blocks; 16 rows each

// ---- CDNA5 async global->LDS helpers (inline asm; bypasses toolchain builtin skew) ----
__device__ __forceinline__ void cdna5_async_load_b128(uint32_t lds_byte_off,
                                                      const void* gaddr) {
#if defined(__gfx1250__)
  // Each lane copies 16 B HBM -> LDS without touching VGPRs; tracked by ASYNCcnt.
  // NT hint: 512 MiB stream is touched exactly once, don't rinse L2 (192 MB) with it.
  asm volatile("global_load_async_to_lds_b128 %0, %1, off th:TH_LOAD_NT"
               :
               : "v"(lds_byte_off), "v"((uint64_t)(uintptr_t)gaddr)
               : "memory");
#else
  (void)lds_byte_off; (void)gaddr;
#endif
}

__device__ __forceinline__ void cdna5_wait_async_le(int n) {
#if defined(__gfx1250__)
  if (n == 0) asm volatile("s_wait_asynccnt 0" ::: "memory");
  else        asm volatile("s_wait_asynccnt 1" ::: "memory");
#else
  (void)n;
#endif
}

// ---- Kernel 1: per-row CE+UL term, per-block deterministic partial sums ----
__global__ __launch_bounds__(LOSS_THREADS) void ce_ul_rows(
    const float* __restrict__ pred,      // [N, C]
    const int*   __restrict__ tgt,       // [N]
    const float* __restrict__ known,     // [N]
    const float* __restrict__ unknown,   // [N]
    float2*      __restrict__ partials,  // [gridDim.x] (term_sum, valid_count)
    int N, int nblk)
{
  __shared__ float buf[2][LOSS_C];            // 8 KB double buffer (one row = 4 KB)
  __shared__ float red[LOSS_THREADS / 32];    // per-wave partials
  __shared__ float sh_xt, sh_max;

  const int tid  = threadIdx.x;
  const int lane = tid & 31;
  const int wid  = tid >> 5;

  // Low 32 bits of a generic shared pointer == LDS byte offset (LDS aperture layout).
  const uint32_t lds0 = (uint32_t)(uintptr_t)&buf[0][4 * tid];
  const uint32_t lds1 = (uint32_t)(uintptr_t)&buf[1][4 * tid];

  float acc_t = 0.0f, acc_v = 0.0f;

  int r = blockIdx.x;
  if (r < N) cdna5_async_load_b128(lds0, pred + (size_t)r * LOSS_C + 4 * tid);
  int p = 0;

  for (; r < N; r += nblk) {
    const int rn = r + nblk;
    if (rn < N) {  // prefetch next row into the other buffer, then wait for current
      cdna5_async_load_b128(p ? lds0 : lds1, pred + (size_t)rn * LOSS_C + 4 * tid);
      cdna5_wait_async_le(1);
    } else {
      cdna5_wait_async_le(0);
    }

    // This wave's own 16 B slice (its own async load is complete per ASYNCcnt).
    const float4 v = *reinterpret_cast<const float4*>(&buf[p][4 * tid]);

    const int t  = tgt[r];
    const int st = (t == IGNORE_INDEX) ? 0 : t;
    if (tid == (st >> 2)) {
      const float* vv = reinterpret_cast<const float*>(&v);
      sh_xt = vv[st & 3];
    }

    // --- row max ---
    float m = fmaxf(fmaxf(v.x, v.y), fmaxf(v.z, v.w));
#pragma unroll
    for (int i = 16; i > 0; i >>= 1) m = fmaxf(m, __shfl_xor(m, i, 32));
    if (lane == 0) red[wid] = m;
    __syncthreads();
    if (tid == 0) {
      float mm = red[0];
#pragma unroll
      for (int w = 1; w < LOSS_THREADS / 32; ++w) mm = fmaxf(mm, red[w]);
      sh_max = mm;
    }
    __syncthreads();
    const float M = sh_max;

    // --- sum exp(x - max) ---
    float s = __expf(v.x - M) + __expf(v.y - M) + __expf(v.z - M) + __expf(v.w - M);
#pragma unroll
    for (int i = 16; i > 0; i >>= 1) s += __shfl_xor(s, i, 32);
    if (lane == 0) red[wid] = s;
    __syncthreads();

    if (tid == 0) {
      float S = 0.0f;
#pragma unroll
      for (int w = 0; w < LOSS_THREADS / 32; ++w) S += red[w];
      const float logZ = M + __logf(S);
      const float logp = sh_xt - logZ;
      const float pe   = __expf(logp);
      const float term = logp * known[r] + __logf(1.0f - pe + 1e-10f) * unknown[r];
      if (t != IGNORE_INDEX) { acc_t += term; acc_v += 1.0f; }
    }
    __syncthreads();  // protect sh_xt/red before next iteration overwrites
    p ^= 1;
  }

  if (tid == 0) partials[blockIdx.x] = make_float2(acc_t, acc_v);
}

// ---- Kernel 2: final tree reduction (deterministic, no float atomics) ----
__global__ __launch_bounds__(256) void ce_ul_final(
    const float2* __restrict__ partials, int nblk, float* __restrict__ out)
{
  __shared__ float2 red[8];
  const int tid = threadIdx.x, lane = tid & 31, wid = tid >> 5;
  float t = 0.0f, v = 0.0f;
  for (int i = tid; i < nblk; i += 256) {
    const float2 pv = partials[i];
    t += pv.x; v += pv.y;
  }
#pragma unroll
  for (int i = 16; i > 0; i >>= 1) {
    t += __shfl_xor(t, i, 32);
    v += __shfl_xor(v, i, 32);
  }
  if (lane == 0) red[wid] = make_float2(t, v);
  __syncthreads();
  if (tid == 0) {
    float T = 0.0f, V = 0.0f;
#pragma unroll
    for (int w = 0; w < 8; ++w) { T += red[w].x; V += red[w].y; }
    out[0] = -T / V;
  }
}

extern "C" void kernel_launch(void* const* d_in, const int* in_sizes, int n_in,
                              void* d_out, int out_size, void* d_ws, size_t ws_size,
                              hipStream_t stream) {
  (void)n_in; (void)out_size;
  const float* pred    = (const float*)d_in[0];
  const int*   tgt     = (const int*)  d_in[1];   // harness delivers integer inputs as int32
  const float* known   = (const float*)d_in[2];
  const float* unknown = (const float*)d_in[3];
  const int N = in_sizes[1];                      // rows; in_sizes[0] == N * 1024

  int nblk = MAX_BLOCKS;
  if (ws_size < (size_t)nblk * sizeof(float2)) {  // shrink partial count if ws is small
    nblk = (int)(ws_size / sizeof(float2));
    if (nblk < 1) nblk = 1;
  }
  float2* partials = (float2*)d_ws;

  ce_ul_rows<<<nblk, LOSS_THREADS, 0, stream>>>(pred, tgt, known, unknown,
                                                partials, N, nblk);
  ce_ul_final<<<1, 256, 0, stream>>>(partials, nblk, (float*)d_out);
}